// UGP_v1_18081812316996
// MI455X (gfx1250) — compile-verified
//
#include <hip/hip_runtime.h>
#include <hip/hip_bf16.h>

#define NS      500000   // n_snps
#define NGENE   20000    // n_genes
#define NNODE   600000   // n_nodes
#define NB      64       // batch
#define NFILT   8
#define HID1    1024
#define HID2    256
#define BN_EPS  1e-5f

typedef float v2f __attribute__((ext_vector_type(2)));
typedef float v8f __attribute__((ext_vector_type(8)));

// ---------------- zero scratch accumulators ----------------
__global__ void k_zero(float* __restrict__ p, int n) {
    int i = blockIdx.x * blockDim.x + threadIdx.x;
    if (i < n) p[i] = 0.0f;
}

// ---------------- w[n] = mean_f filters[f, n] ----------------
__global__ void k_filter_mean(const float* __restrict__ filters, float* __restrict__ w) {
    int n = blockIdx.x * blockDim.x + threadIdx.x;
    if (n >= NS) return;
    float s = 0.0f;
#pragma unroll
    for (int f = 0; f < NFILT; ++f) s += filters[(size_t)f * NS + n];
    w[n] = s * (1.0f / NFILT);
}

// ---------------- filters passthrough (second tuple output), 16B vectors ----------------
__global__ void k_copy4(const float4* __restrict__ src, float4* __restrict__ dst, int n4) {
    int i = blockIdx.x * blockDim.x + threadIdx.x;
    if (i < n4) dst[i] = src[i];
}

// ---------------- gather snp by snp_ids, scale by w, segment-sum into gene ----------------
__global__ void k_scatter(const float* __restrict__ snp,
                          const int*   __restrict__ snp_ids,
                          const int*   __restrict__ node_gene,
                          const float* __restrict__ w,
                          float*       __restrict__ gene) {
    int idx = blockIdx.x * blockDim.x + threadIdx.x;
    const int total = NB * NNODE;
    if (idx >= total) return;
    int b = idx / NNODE;
    int j = idx - b * NNODE;          // consecutive lanes -> consecutive j: index loads coalesce
    int sid = snp_ids[j];
    int g   = node_gene[j];
    float v = snp[(size_t)b * NS + sid] * w[sid];   // random 4B gather, L2-resident
    atomicAdd(&gene[b * NGENE + g], v);
}

// ---------------- fp32 WMMA GEMM: C[64,N] += A[64,K] * B[K,N], K-chunked ----------------
// grid.x = N/16 (n-tile), grid.y = K/kChunk (k-chunk), block = 32 threads (one wave).
// One wave owns all 4 M-tiles for its n-tile so the B fragment (weight stream) is
// loaded exactly once per k-step. All addresses are strength-reduced to pointer
// increments so the hot loop is loads + 4x v_wmma + 6x 64-bit pointer adds.
__global__ void k_gemm_wmma_f32(const float* __restrict__ A,
                                const float* __restrict__ B,
                                float*       __restrict__ C,
                                int N, int K, int kChunk) {
    const int lane = threadIdx.x;        // 0..31, full wave, no divergence in loop
    const int half = lane >> 4;          // 0 or 1
    const int r    = lane & 15;
    const int n0   = blockIdx.x * 16;
    const int k0   = blockIdx.y * kChunk;
    const int ka0  = k0 + 2 * half;      // this half-wave's first K pair (even -> 8B aligned)

    v8f acc0 = {}, acc1 = {}, acc2 = {}, acc3 = {};

    // A row pointers (M tiles 0..3), stepped by 4 floats per iteration
    const float* ap0 = A + (size_t)(r      ) * K + ka0;
    const float* ap1 = A + (size_t)(r + 16 ) * K + ka0;
    const float* ap2 = A + (size_t)(r + 32 ) * K + ka0;
    const float* ap3 = A + (size_t)(r + 48 ) * K + ka0;
    // B row pointers (rows ka, ka+1), stepped by 4*N floats per iteration
    const float* bp0 = B + (size_t)ka0 * N + n0 + r;
    const float* bp1 = bp0 + N;
    const size_t bstep = (size_t)4 * N;
    const size_t bpf   = (size_t)64 * N;   // stream-ahead distance (64 K-rows)

    const int steps = kChunk >> 2;
    for (int s = 0; s < steps; ++s) {
        v2f bf; bf.x = *bp0; bf.y = *bp1;
        __builtin_prefetch(bp0 + bpf, 0, 1);    // speculative: OOB prefetch is dropped

        v2f a0 = *(const v2f*)ap0;              // contiguous K pair -> global_load_b64
        v2f a1 = *(const v2f*)ap1;
        v2f a2 = *(const v2f*)ap2;
        v2f a3 = *(const v2f*)ap3;

        acc0 = __builtin_amdgcn_wmma_f32_16x16x4_f32(false, a0, false, bf, (short)0, acc0, false, false);
        acc1 = __builtin_amdgcn_wmma_f32_16x16x4_f32(false, a1, false, bf, (short)0, acc1, false, false);
        acc2 = __builtin_amdgcn_wmma_f32_16x16x4_f32(false, a2, false, bf, (short)0, acc2, false, false);
        acc3 = __builtin_amdgcn_wmma_f32_16x16x4_f32(false, a3, false, bf, (short)0, acc3, false, false);

        ap0 += 4; ap1 += 4; ap2 += 4; ap3 += 4;
        bp0 += bstep; bp1 += bstep;
    }

    // C/D layout: VGPR i -> row i (lanes 0-15) / row i+8 (lanes 16-31), col = n0+r.
    // Base pointer walk; the 16N/32N/48N constants fold into the 24-bit ioffset.
    float* cp = C + (size_t)(8 * half) * N + n0 + r;
#pragma unroll
    for (int i = 0; i < 8; ++i) {
        atomicAdd(cp,                    acc0[i]);
        atomicAdd(cp + (size_t)16 * N,   acc1[i]);
        atomicAdd(cp + (size_t)32 * N,   acc2[i]);
        atomicAdd(cp + (size_t)48 * N,   acc3[i]);
        cp += N;
    }
}

// ---------------- training-mode BatchNorm (batch stats over B=64, biased var) + ReLU ----------------
__global__ void k_bn_relu(const float* __restrict__ z,
                          const float* __restrict__ bias,
                          const float* __restrict__ gamma,
                          const float* __restrict__ beta,
                          float* __restrict__ h, int N) {
    int c = blockIdx.x * blockDim.x + threadIdx.x;  // one column per thread; row-major -> coalesced
    if (c >= N) return;
    const float bb = bias[c];
    float s = 0.0f, sq = 0.0f;
    for (int rr = 0; rr < NB; ++rr) {
        float x = z[rr * N + c] + bb;
        s += x; sq += x * x;
    }
    const float m   = s * (1.0f / NB);
    const float var = sq * (1.0f / NB) - m * m;
    const float inv = rsqrtf(var + BN_EPS);
    const float g = gamma[c], bt = beta[c];
    for (int rr = 0; rr < NB; ++rr) {
        float x = z[rr * N + c] + bb;
        float y = g * (x - m) * inv + bt;
        h[rr * N + c] = fmaxf(y, 0.0f);
    }
}

// ---------------- preds[b] = h2[b,:] . W3 + b3 ----------------
__global__ void k_final(const float* __restrict__ h2,
                        const float* __restrict__ W3,
                        const float* __restrict__ b3,
                        float* __restrict__ out) {
    int b = threadIdx.x;
    if (b >= NB) return;
    float s = b3[0];
    for (int c = 0; c < HID2; ++c) s += h2[b * HID2 + c] * W3[c];
    out[b] = s;
}

extern "C" void kernel_launch(void* const* d_in, const int* in_sizes, int n_in,
                              void* d_out, int out_size, void* d_ws, size_t ws_size,
                              hipStream_t stream) {
    const float* snp       = (const float*)d_in[0];
    const int*   snp_ids   = (const int*)  d_in[1];
    const int*   node_gene = (const int*)  d_in[2];
    const float* filters   = (const float*)d_in[3];
    const float* W1        = (const float*)d_in[4];
    const float* b1        = (const float*)d_in[5];
    const float* g1        = (const float*)d_in[6];
    const float* beta1     = (const float*)d_in[7];
    const float* W2        = (const float*)d_in[8];
    const float* b2        = (const float*)d_in[9];
    const float* g2        = (const float*)d_in[10];
    const float* beta2     = (const float*)d_in[11];
    const float* W3        = (const float*)d_in[12];
    const float* b3        = (const float*)d_in[13];
    float* out = (float*)d_out;

    // scratch layout (floats): gene | z1 | z2 | h1 | h2 | w   (~7.8 MB)
    float* ws   = (float*)d_ws;
    float* gene = ws;                         // 64*20000
    float* z1   = gene + NB * NGENE;          // 64*1024
    float* z2   = z1   + NB * HID1;           // 64*256
    float* h1   = z2   + NB * HID2;           // 64*1024
    float* h2   = h1   + NB * HID1;           // 64*256
    float* w    = h2   + NB * HID2;           // 500000

    // 1) zero the atomic accumulators (gene, z1, z2 are contiguous)
    const int zn = NB * NGENE + NB * HID1 + NB * HID2;
    k_zero<<<(zn + 255) / 256, 256, 0, stream>>>(gene, zn);

    // 2) per-SNP fused weight + filters passthrough output
    k_filter_mean<<<(NS + 255) / 256, 256, 0, stream>>>(filters, w);
    const int n4 = (NFILT * NS) / 4;
    k_copy4<<<(n4 + 255) / 256, 256, 0, stream>>>((const float4*)filters,
                                                  (float4*)(out + NB), n4);

    // 3) gather-scale-scatter into gene[64, 20000]
    const int total = NB * NNODE;
    k_scatter<<<(total + 255) / 256, 256, 0, stream>>>(snp, snp_ids, node_gene, w, gene);

    // 4) GEMM1: gene[64,20000] x W1[20000,1024] (K split 20 x 1000)
    dim3 gg1(HID1 / 16, 20);
    k_gemm_wmma_f32<<<gg1, 32, 0, stream>>>(gene, W1, z1, HID1, NGENE, 1000);

    // 5) BN1 + ReLU
    k_bn_relu<<<(HID1 + 255) / 256, 256, 0, stream>>>(z1, b1, g1, beta1, h1, HID1);

    // 6) GEMM2: h1[64,1024] x W2[1024,256] (K split 4 x 256)
    dim3 gg2(HID2 / 16, 4);
    k_gemm_wmma_f32<<<gg2, 32, 0, stream>>>(h1, W2, z2, HID2, HID1, 256);

    // 7) BN2 + ReLU
    k_bn_relu<<<(HID2 + 255) / 256, 256, 0, stream>>>(z2, b2, g2, beta2, h2, HID2);

    // 8) final head -> preds[64]
    k_final<<<1, 64, 0, stream>>>(h2, W3, b3, out);
}